// NuFACoef_15496242004659
// MI455X (gfx1250) — compile-verified
//
#include <hip/hip_runtime.h>
#include <hip/hip_bf16.h>

// CDNA5 / gfx1250. wave32 only.
typedef __attribute__((ext_vector_type(2))) float    v2f;
typedef __attribute__((ext_vector_type(8))) float    v8f;
typedef __attribute__((ext_vector_type(4))) unsigned u32x4;
typedef __attribute__((ext_vector_type(4))) int      i32x4;
typedef __attribute__((ext_vector_type(8))) int      i32x8;

#define WPB 4              // waves per block
#define TILE_FLOATS 4096   // 64x64 fp32
#define TILE_BYTES  16384

// Issue one TDM load of a 64x64 fp32 tile (row-major, stride 64) into LDS.
// D# per CDNA5 ISA ch.8: group0 {count=1, lds_addr, global_addr, type=2},
// group1 {data_size=4B, tensor_dim0=64, tensor_dim1=64, tile_dim0=64,
// tile_dim1=64, tensor_dim0_stride=64}. 2D tensor -> groups 2/3 zero.
static __device__ inline void tdm_load_tile(const float* g, unsigned lds_byte) {
  unsigned long long ga = (unsigned long long)g;
  unsigned lo = (unsigned)__builtin_amdgcn_readfirstlane((int)(unsigned)ga);
  unsigned hi = (unsigned)__builtin_amdgcn_readfirstlane((int)(unsigned)(ga >> 32));
  unsigned la = (unsigned)__builtin_amdgcn_readfirstlane((int)lds_byte);
  u32x4 g0 = { 1u, la, lo, (hi & 0x01FFFFFFu) | 0x80000000u };
  i32x8 g1 = { (int)(2u << 16),        // data_size = 4 bytes
               (int)(64u << 16),       // tensor_dim0[15:0] = 64
               (int)(64u << 16),       // tensor_dim1[15:0] = 64
               (int)(64u << 16),       // tile_dim0 = 64
               64,                     // tile_dim1 = 64, tile_dim2 = 0
               64,                     // tensor_dim0_stride = 64 elements
               0, 0 };
  i32x4 g2 = { 0, 0, 0, 0 };
  i32x4 g3 = { 0, 0, 0, 0 };
#if __clang_major__ >= 23
  i32x8 g4 = { 0, 0, 0, 0, 0, 0, 0, 0 };
  __builtin_amdgcn_tensor_load_to_lds(g0, g1, g2, g3, g4, 0);
#else
  __builtin_amdgcn_tensor_load_to_lds(g0, g1, g2, g3, 0);
#endif
}

// Chain c <- c @ X five times starting from c = ones(64) with
// V_WMMA_F32_16X16X4_F32, X held register-resident in B-matrix layout
// (64 v2f pairs = 128 VGPRs), then accumulate weighted trace sums of c2..c5.
static __device__ inline float chain_compute(const float* __restrict__ xb,
                                             int lane,
                                             const float* __restrict__ w) {
  const int    half = (lane & 16) ? 2 : 0;   // K offset for this half-wave
  const float* colp = xb + (lane & 15);      // column within a 16-wide n-tile

  // One-shot LDS -> VGPR load of X in B(4x16)-tile layout.
  // Pair (row r, row r+1) at the same column -> ds_load_2addr offset1 = +64.
  v2f xr[64];
#pragma unroll
  for (int kt = 0; kt < 16; ++kt) {
    const float* rp = colp + (4 * kt + half) * 64;
#pragma unroll
    for (int nt = 0; nt < 4; ++nt) {
      v2f t; t.x = rp[nt * 16]; t.y = rp[nt * 16 + 64];
      xr[kt * 4 + nt] = t;
    }
  }

  float c0 = 1.0f, c1 = 1.0f, c2 = 1.0f, c3 = 1.0f;  // c chunks nt=0..3
  float outacc = 0.0f;

#pragma unroll
  for (int it = 0; it < 5; ++it) {
    v8f acc0 = {0,0,0,0,0,0,0,0};
    v8f acc1 = {0,0,0,0,0,0,0,0};
    v8f acc2 = {0,0,0,0,0,0,0,0};
    v8f acc3 = {0,0,0,0,0,0,0,0};
#pragma unroll
    for (int kt = 0; kt < 16; ++kt) {
      v2f A;
      if (it == 0) {                 // c = ones: no broadcast needed
        A.x = 1.0f; A.y = 1.0f;
      } else {
        const int   r  = kt >> 2;
        const int   sb = ((kt & 3) << 2) + half;
        const float cr = (r == 0) ? c0 : (r == 1) ? c1 : (r == 2) ? c2 : c3;
        A.x = __shfl(cr, sb);        // c[4kt + half]
        A.y = __shfl(cr, sb + 1);    // c[4kt + half + 1]
      }
      acc0 = __builtin_amdgcn_wmma_f32_16x16x4_f32(false, A, false, xr[kt*4+0], (short)0, acc0, false, false);
      acc1 = __builtin_amdgcn_wmma_f32_16x16x4_f32(false, A, false, xr[kt*4+1], (short)0, acc1, false, false);
      acc2 = __builtin_amdgcn_wmma_f32_16x16x4_f32(false, A, false, xr[kt*4+2], (short)0, acc2, false, false);
      acc3 = __builtin_amdgcn_wmma_f32_16x16x4_f32(false, A, false, xr[kt*4+3], (short)0, acc3, false, false);
    }
    // Every row of D is identical; VGPR0 carries the new c chunk on all lanes.
    c0 = acc0[0]; c1 = acc1[0]; c2 = acc2[0]; c3 = acc3[0];

    if (it >= 1) {                   // c = colsum(x^(it+1)); collect x^2..x^5
      float s1 = 0.0f, s2 = 0.0f, s3 = 0.0f, s4 = 0.0f;
      float vv[4] = { c0, c1, c2, c3 };
#pragma unroll
      for (int q = 0; q < 4; ++q) {
        const float v = vv[q], v2 = v * v;
        s1 += v; s2 += v2; s3 += v2 * v; s4 += v2 * v2;
      }
      // Lanes 0-15 already cover all 64 k-values (lanes 16-31 mirror them):
      // butterfly within the 16-lane group only.
#pragma unroll
      for (int off = 8; off > 0; off >>= 1) {
        s1 += __shfl_xor(s1, off);
        s2 += __shfl_xor(s2, off);
        s3 += __shfl_xor(s3, off);
        s4 += __shfl_xor(s4, off);
      }
      const int i = it - 1;          // power-row index 0..3
      outacc += w[4*i+0]*s1 + w[4*i+1]*s2 + w[4*i+2]*s3 + w[4*i+3]*s4;
    }
  }
  return outacc;
}

__global__ __launch_bounds__(WPB * 32)
void nufa_coef_kernel(const float* __restrict__ x,
                      const float* __restrict__ coef,
                      float* __restrict__ out,
                      int Btot, int per) {
  extern __shared__ float smem[];
  const int lane = (int)(threadIdx.x & 31u);
  const int wib  = (int)(threadIdx.x >> 5);
  const long long gw = (long long)blockIdx.x * WPB + wib;
  const long long b0 = gw * per;
  if (b0 >= Btot) return;
  int nb = (int)(Btot - b0); if (nb > per) nb = per;

  // Per-lane precomputed weights w[i][j] = coef[i][j] / 16^(i+j+2).
  float w[16];
  {
    float rowbase = 1.0f / 256.0f;               // 16^-2
#pragma unroll
    for (int i = 0; i < 4; ++i) {
      float p = rowbase;
#pragma unroll
      for (int j = 0; j < 4; ++j) { w[4*i+j] = coef[4*i+j] * p; p *= 0.0625f; }
      rowbase *= 0.0625f;
    }
  }

  // Per-wave double buffers. Dynamic LDS begins after static LDS.
  const unsigned ldsBase =
      (unsigned)__builtin_amdgcn_groupstaticsize() + (unsigned)(wib * 2 * TILE_BYTES);
  const float* wbuf = smem + (size_t)wib * 2 * TILE_FLOATS;

  tdm_load_tile(x + (unsigned long long)b0 * TILE_FLOATS, ldsBase);
  for (int t = 0; t < nb; ++t) {
    if (t + 1 < nb) {
      tdm_load_tile(x + (unsigned long long)(b0 + t + 1) * TILE_FLOATS,
                    ldsBase + (unsigned)(((t + 1) & 1) * TILE_BYTES));
      __builtin_amdgcn_s_wait_tensorcnt(1);   // oldest (current) tile resident
    } else {
      __builtin_amdgcn_s_wait_tensorcnt(0);
    }
    const float* xb = wbuf + (size_t)(t & 1) * TILE_FLOATS;
    const float r = chain_compute(xb, lane, w);
    if (lane == 0) out[b0 + t] = r;
  }
}

extern "C" void kernel_launch(void* const* d_in, const int* in_sizes, int n_in,
                              void* d_out, int out_size, void* d_ws, size_t ws_size,
                              hipStream_t stream) {
  const float* x    = (const float*)d_in[0];
  const float* coef = (const float*)d_in[1];
  float* out        = (float*)d_out;
  const int Btot = in_sizes[0] / (64 * 64);
  const int per = 4;                               // batches per wave
  const int batchesPerBlock = WPB * per;
  const int blocks = (Btot + batchesPerBlock - 1) / batchesPerBlock;
  const size_t shmem = (size_t)WPB * 2 * TILE_BYTES;   // 128 KB of the 320 KB WGP LDS
  nufa_coef_kernel<<<blocks, WPB * 32, shmem, stream>>>(x, coef, out, Btot, per);
}